// UDAGCN_Encoder_609885356320
// MI455X (gfx1250) — compile-verified
//
#include <hip/hip_runtime.h>
#include <hip/hip_bf16.h>

typedef float v2f __attribute__((ext_vector_type(2)));
typedef float v8f __attribute__((ext_vector_type(8)));

#define D 128
#define KSTEPS 32   // 128 / 4

// ---------------------------------------------------------------------------
// Degree / norm kernels
// ---------------------------------------------------------------------------
__global__ void k_deg_init(float* __restrict__ deg, int n) {
    int i = blockIdx.x * blockDim.x + threadIdx.x;
    if (i < n) deg[i] = 1.0f;  // self loop contributes 1
}

__global__ void k_deg_count(const int* __restrict__ row, float* __restrict__ deg, int nE) {
    int e = blockIdx.x * blockDim.x + threadIdx.x;
    if (e < nE) atomicAdd(&deg[row[e]], 1.0f);
}

__global__ void k_dinv(const float* __restrict__ deg, float* __restrict__ dinv, int n) {
    int i = blockIdx.x * blockDim.x + threadIdx.x;
    if (i < n) {
        float d = deg[i];
        dinv[i] = (d > 0.0f) ? rsqrtf(d) : 0.0f;
    }
}

// ---------------------------------------------------------------------------
// Pack W[128,128] row-major into Wp[(k/2)*128 + c] = {W[k][c], W[k+1][c]}
// so the GEMM's B-operand fetch is a single contiguous b64 load.
// ---------------------------------------------------------------------------
__global__ void k_packW(const float* __restrict__ W, float2* __restrict__ Wp) {
    int idx = blockIdx.x * blockDim.x + threadIdx.x;  // 0 .. 64*128-1
    if (idx < 64 * D) {
        int kh = idx >> 7;   // k/2
        int c  = idx & 127;
        float2 v;
        v.x = W[(size_t)(kh * 2) * D + c];
        v.y = W[(size_t)(kh * 2 + 1) * D + c];
        Wp[idx] = v;
    }
}

// ---------------------------------------------------------------------------
// Dense transform: H[N,128] = A[N,128] @ W[128,128] using V_WMMA_F32_16X16X4_F32
// Block = 256 threads = 8 waves = one 16-row x 128-col stripe; grid = ceil(N/16).
// A tile (16x128 = 8KB) staged in LDS once per block, shared by all 8 waves.
//
// f32 WMMA layouts (ISA 7.12.2):
//   A 16x4 : lanes 0-15 -> M=lane, {v0,v1} = K{0,1}; lanes 16-31 -> K{2,3}
//   B 4x16 : lanes 0-15 -> N=lane, {v0,v1} = K{0,1}; lanes 16-31 -> K{2,3}
//   C/D    : vgpr r, lanes 0-15 -> M=r, N=lane; lanes 16-31 -> M=8+r, N=lane-16
// ---------------------------------------------------------------------------
__global__ void __launch_bounds__(256)
k_gemm_wmma(const float* __restrict__ A, const float2* __restrict__ Wp,
            float* __restrict__ H, int nRows) {
    __shared__ __align__(16) float sA[16 * D];   // 8 KB A tile

    const int rt   = blockIdx.x;           // row tile
    const int tid  = threadIdx.x;
    const int wave = tid >> 5;             // col tile 0..7
    const int lane = tid & 31;
    const int m    = lane & 15;
    const int hs   = lane >> 4;            // half select (K pair 0/1 vs 2/3)

    // Cooperative A-tile load: 2048 floats = 256 threads x 8 (two b128 each).
    {
        const int base  = tid * 8;                 // 0..2047
        int arow        = rt * 16 + (base >> 7);
        const int arowC = (arow < nRows) ? arow : (nRows - 1);  // clamp
        const int acol  = base & 127;
        const float4* src = (const float4*)(A + (size_t)arowC * D + acol);
        float4 p0 = src[0];
        float4 p1 = src[1];
        ((float4*)(sA + base))[0] = p0;
        ((float4*)(sA + base))[1] = p1;
    }
    __syncthreads();

    const float* aRow   = sA + m * D;
    const int   colBase = wave * 16 + m;

    v8f acc = {};
#pragma unroll 8
    for (int kk = 0; kk < KSTEPS; ++kk) {
        const int k0 = kk * 4 + hs * 2;
        v2f a = *(const v2f*)(aRow + k0);                       // ds_load_b64
        float2 bp = Wp[(size_t)(kk * 2 + hs) * D + colBase];    // global b64
        v2f b; b.x = bp.x; b.y = bp.y;
        acc = __builtin_amdgcn_wmma_f32_16x16x4_f32(
            /*neg_a=*/false, a, /*neg_b=*/false, b,
            /*c_mod=*/(short)0, acc, /*reuse_a=*/false, /*reuse_b=*/false);
    }

    // Epilogue: tile-uniform bounds check -> straight-line stores normally.
    if (rt * 16 + 16 <= nRows) {
#pragma unroll
        for (int r = 0; r < 8; ++r)
            H[(size_t)(rt * 16 + hs * 8 + r) * D + colBase] = acc[r];
    } else {
        for (int r = 0; r < 8; ++r) {
            const int orow = rt * 16 + hs * 8 + r;
            if (orow < nRows)
                H[(size_t)orow * D + colBase] = acc[r];
        }
    }
}

// ---------------------------------------------------------------------------
// Per-node init: out[i,:] = dinv[i]^2 * h[i,:] + bias   (self-loop + bias,
// single writer per node -> no atomics). Block = 128 threads = one node row.
// ---------------------------------------------------------------------------
__global__ void __launch_bounds__(128)
k_self_bias(const float* __restrict__ H, const float* __restrict__ dinv,
            const float* __restrict__ bias, float* __restrict__ out, int n) {
    const int i = blockIdx.x;
    const int c = threadIdx.x;
    if (i < n) {
        const float s = dinv[i];
        out[(size_t)i * D + c] = s * s * H[(size_t)i * D + c] + bias[c];
    }
}

// ---------------------------------------------------------------------------
// Edge scatter: one wave per edge, lane handles 4 channels (32*4 = 128).
// out[col] += dinv[row]*dinv[col] * h[row]; grid-stride + prefetch of the
// next gathered row (global_prefetch_b8). h is L2-resident (25.6MB << 192MB).
// ---------------------------------------------------------------------------
__global__ void __launch_bounds__(256)
k_edge_scatter(const float* __restrict__ H, const float* __restrict__ dinv,
               const int* __restrict__ row, const int* __restrict__ col,
               float* __restrict__ out, int nE) {
    const int lane  = threadIdx.x & 31;
    const int wid   = blockIdx.x * (blockDim.x >> 5) + (threadIdx.x >> 5);
    const int nwave = gridDim.x * (blockDim.x >> 5);

    for (int e = wid; e < nE; e += nwave) {
        const int r  = row[e];
        const int c  = col[e];
        const int en = e + nwave;
        if (en < nE)
            __builtin_prefetch(&H[(size_t)row[en] * D + lane * 4], 0, 0);

        const float nrm = dinv[r] * dinv[c];
        const float4 hv = ((const float4*)(H + (size_t)r * D))[lane];
        float* o = out + (size_t)c * D + lane * 4;
        atomicAdd(o + 0, nrm * hv.x);
        atomicAdd(o + 1, nrm * hv.y);
        atomicAdd(o + 2, nrm * hv.z);
        atomicAdd(o + 3, nrm * hv.w);
    }
}

__global__ void k_relu(float* __restrict__ t, size_t n) {
    size_t i = (size_t)blockIdx.x * blockDim.x + threadIdx.x;
    if (i < n) t[i] = fmaxf(t[i], 0.0f);
}

// ---------------------------------------------------------------------------
// Host-side launch
// ---------------------------------------------------------------------------
extern "C" void kernel_launch(void* const* d_in, const int* in_sizes, int n_in,
                              void* d_out, int out_size, void* d_ws, size_t ws_size,
                              hipStream_t stream) {
    const float* x   = (const float*)d_in[0];
    const int*   ei  = (const int*)d_in[1];   // [2, E] (int32 under default JAX x64-off)
    const float* W1  = (const float*)d_in[2];
    const float* b1  = (const float*)d_in[3];
    const float* W2  = (const float*)d_in[4];
    const float* b2  = (const float*)d_in[5];
    float* outF      = (float*)d_out;

    const int N = in_sizes[0] / D;
    const int E = in_sizes[1] / 2;
    const int* row = ei;
    const int* col = ei + E;

    // workspace layout (floats): h[N*D] | t[N*D] | deg[N] | dinv[N] | Wp[128*128]
    float*  ws   = (float*)d_ws;
    float*  h    = ws;
    float*  t    = ws + (size_t)N * D;
    float*  deg  = ws + 2 * (size_t)N * D;
    float*  dinv = deg + N;
    float2* Wp   = (float2*)(dinv + N);

    const int rowTiles = (N + 15) / 16;
    const int edgeGrid = 2048;           // grid-stride over edges, 8 waves/block
    const int packGrid = (64 * D + 255) / 256;

    // ---- GCN norm -------------------------------------------------------
    k_deg_init<<<(N + 255) / 256, 256, 0, stream>>>(deg, N);
    k_deg_count<<<(E + 255) / 256, 256, 0, stream>>>(row, deg, E);
    k_dinv<<<(N + 255) / 256, 256, 0, stream>>>(deg, dinv, N);

    // ---- Layer 1: t = scatter(norm * (x@W1)) + b1 ; ReLU ---------------
    k_packW<<<packGrid, 256, 0, stream>>>(W1, Wp);
    k_gemm_wmma<<<rowTiles, 256, 0, stream>>>(x, Wp, h, N);
    k_self_bias<<<N, 128, 0, stream>>>(h, dinv, b1, t, N);
    k_edge_scatter<<<edgeGrid, 256, 0, stream>>>(h, dinv, row, col, t, E);
    k_relu<<<(int)(((size_t)N * D + 255) / 256), 256, 0, stream>>>(t, (size_t)N * D);

    // ---- Layer 2: d_out = scatter(norm * (t@W2)) + b2 -------------------
    k_packW<<<packGrid, 256, 0, stream>>>(W2, Wp);
    k_gemm_wmma<<<rowTiles, 256, 0, stream>>>(t, Wp, h, N);
    k_self_bias<<<N, 128, 0, stream>>>(h, dinv, b2, outF, N);
    k_edge_scatter<<<edgeGrid, 256, 0, stream>>>(h, dinv, row, col, outF, E);
}